// HelperPointLoss_11562051961325
// MI455X (gfx1250) — compile-verified
//
#include <hip/hip_runtime.h>

typedef float v2f __attribute__((ext_vector_type(2)));
typedef float v8f __attribute__((ext_vector_type(8)));

#define B_ 16
#define N_ 4096
#define M_ 4096
#define INF_I 0x7F800000   // +inf bits; > bits of every finite non-negative float,
                           // and every negative float is < 0 as signed int.

static __device__ __forceinline__ int imin(int a, int b) { return a < b ? a : b; }

// ---------------------------------------------------------------------------
// Pass 1: pack pred -> [x,y,z,||x||^2], gt -> [-2x,-2y,-2z,||y||^2]; init colmin
// ---------------------------------------------------------------------------
__global__ __launch_bounds__(256) void cd_pack_init(const float* __restrict__ pred,
                                                    const float* __restrict__ gt,
                                                    float4* __restrict__ xp,
                                                    float4* __restrict__ gp,
                                                    int* __restrict__ colmin)
{
    int i = blockIdx.x * blockDim.x + threadIdx.x;
    if (i < B_ * N_) {
        float x = pred[i * 3 + 0], y = pred[i * 3 + 1], z = pred[i * 3 + 2];
        xp[i] = make_float4(x, y, z, x * x + y * y + z * z);
    }
    if (i < B_ * M_) {
        float x = gt[i * 3 + 0], y = gt[i * 3 + 1], z = gt[i * 3 + 2];
        gp[i] = make_float4(-2.0f * x, -2.0f * y, -2.0f * z, x * x + y * y + z * z);
        colmin[i] = INF_I;
    }
}

// ---------------------------------------------------------------------------
// Pass 2: pairwise sq-distances via V_WMMA_F32_16X16X4_F32.
// Wave owns 32 pred rows (two A fragments); each loaded B tile + C splat feeds
// two WMMAs. Workgroup = 8 waves = 256 rows; grid = B * N/256 = 256 WGs.
// All minima tracked as signed-int float bits (branchless, canonicalize-free).
// ---------------------------------------------------------------------------
__global__ __launch_bounds__(256) void cd_tiles(const float4* __restrict__ xp,
                                                const float4* __restrict__ gp,
                                                int* __restrict__ rowmin,
                                                int* __restrict__ colmin)
{
    __shared__ int scol[M_];                       // 16 KB of 320 KB WGP LDS
    const int tid = threadIdx.x;
    for (int j = tid; j < M_; j += 256) scol[j] = INF_I;
    __syncthreads();

    const int b    = blockIdx.x >> 4;              // 16 row-blocks per batch
    const int nb   = blockIdx.x & 15;
    const int wave = tid >> 5;
    const int lane = tid & 31;
    const int l15  = lane & 15;
    const bool hi  = lane >= 16;

    const int n0 = nb * 256 + wave * 32;

    // Two A fragments (16x4 f32): lanes 0-15 hold K={0,1}, lanes 16-31 K={2,3}
    float4 xv0 = xp[b * N_ + n0 + l15];
    float4 xv1 = xp[b * N_ + n0 + 16 + l15];
    v2f a0, a1;
    a0.x = hi ? xv0.z : xv0.x;  a0.y = hi ? xv0.w : xv0.y;   // K2:K0 | K3(=||x||^2):K1
    a1.x = hi ? xv1.z : xv1.x;  a1.y = hi ? xv1.w : xv1.y;

    int dmin[16];
#pragma unroll
    for (int i = 0; i < 16; ++i) dmin[i] = INF_I;

    const float4* gb = gp + (size_t)b * M_;
    for (int mt = 0; mt < M_ / 16; mt += 2) {
        // Issue both B-tile loads up front: second hides behind first's compute
        float4 y0 = gb[mt * 16 + l15];
        float4 y1 = gb[mt * 16 + 16 + l15];

#pragma unroll
        for (int u = 0; u < 2; ++u) {
            float4 yv = u ? y1 : y0;
            // B fragment (4x16): VGPR0 = rows K0|K1, VGPR1 = rows K2|K3
            v2f bb;
            bb.x = hi ? yv.y : yv.x;               // K1 : K0   (-2y components)
            bb.y = hi ? 1.0f : yv.z;               // K3(=1) : K2
            v8f c;
#pragma unroll
            for (int i = 0; i < 8; ++i) c[i] = yv.w;   // C[m][n] = ||y_n||^2

            // D[m][n] = ||x_m||^2 - 2 x.y + ||y_n||^2   (squared distance)
            v8f d0 = __builtin_amdgcn_wmma_f32_16x16x4_f32(
                false, a0, false, bb, (short)0, c, false, false);
            v8f d1 = __builtin_amdgcn_wmma_f32_16x16x4_f32(
                false, a1, false, bb, (short)0, c, false, false);

            // Column-min tree seeded from the first pair (no INF literal min)
            int t = imin(__float_as_int(d0[0]), __float_as_int(d1[0]));
            dmin[0] = imin(dmin[0], __float_as_int(d0[0]));
            dmin[8] = imin(dmin[8], __float_as_int(d1[0]));
#pragma unroll
            for (int i = 1; i < 8; ++i) {
                int u0 = __float_as_int(d0[i]);
                int u1 = __float_as_int(d1[i]);
                dmin[i]     = imin(dmin[i], u0);
                dmin[8 + i] = imin(dmin[8 + i], u1);
                t = imin(t, imin(u0, u1));
            }
            // Both lane halves hit the same column entry -> covers all 32 rows,
            // no shuffle, no divergence; EXEC stays all-ones for the WMMAs.
            atomicMin(&scol[(mt + u) * 16 + l15], t);
        }
    }

    // Row minima: integer-min reduce across the 16 lanes of each half-wave
#pragma unroll
    for (int i = 0; i < 16; ++i) {
        int v = dmin[i];
        v = imin(v, __shfl_xor(v, 1));
        v = imin(v, __shfl_xor(v, 2));
        v = imin(v, __shfl_xor(v, 4));
        v = imin(v, __shfl_xor(v, 8));
        if (l15 == 0) {
            int row = n0 + ((i >= 8) ? 16 : 0) + (i & 7) + (hi ? 8 : 0);
            rowmin[b * N_ + row] = v;              // sq-domain bits; decode in pass 3
        }
    }

    __syncthreads();
    for (int j = tid; j < M_; j += 256)
        atomicMin(&colmin[b * M_ + j], scol[j]);
}

// ---------------------------------------------------------------------------
// Pass 3: decode bits -> clamped sq-dist, sqrt, means, 3 scalar outputs
// ---------------------------------------------------------------------------
static __device__ __forceinline__ float decode_sq(int v)
{
    return (v < 0) ? 0.0f : __int_as_float(v);     // clamp(sq, 0) via sign bit
}

__global__ __launch_bounds__(256) void cd_reduce(const int* __restrict__ rowmin,
                                                 const int* __restrict__ colmin,
                                                 float* __restrict__ out)
{
    __shared__ float sr[256], sc[256];
    int tid = threadIdx.x;
    float s1 = 0.0f, s2 = 0.0f;
    for (int i = tid; i < B_ * N_; i += 256) s1 += sqrtf(decode_sq(rowmin[i]));
    for (int i = tid; i < B_ * M_; i += 256) s2 += sqrtf(decode_sq(colmin[i]));
    sr[tid] = s1; sc[tid] = s2;
    __syncthreads();
    for (int s = 128; s > 0; s >>= 1) {
        if (tid < s) { sr[tid] += sr[tid + s]; sc[tid] += sc[tid + s]; }
        __syncthreads();
    }
    if (tid == 0) {
        float loss = sr[0] / (float)(B_ * N_) + sc[0] / (float)(B_ * M_);
        out[0] = 1.0f * loss;      // WEIGHT * loss
        out[1] = loss;             // helper_loss
        out[2] = loss * 0.1f;      // cderr = p=1 chamfer * XYZ_UNIT (P==1)
    }
}

// ---------------------------------------------------------------------------
extern "C" void kernel_launch(void* const* d_in, const int* in_sizes, int n_in,
                              void* d_out, int out_size, void* d_ws, size_t ws_size,
                              hipStream_t stream)
{
    const float* pred = (const float*)d_in[0];
    const float* gt   = (const float*)d_in[1];
    float* out = (float*)d_out;

    char* ws = (char*)d_ws;
    float4* xp = (float4*)ws;      ws += (size_t)B_ * N_ * sizeof(float4);
    float4* gp = (float4*)ws;      ws += (size_t)B_ * M_ * sizeof(float4);
    int* rowmin = (int*)ws;        ws += (size_t)B_ * N_ * sizeof(int);
    int* colmin = (int*)ws;        // B_*M_ ints

    const int total = B_ * N_;     // == B_*M_
    cd_pack_init<<<(total + 255) / 256, 256, 0, stream>>>(pred, gt, xp, gp, colmin);
    cd_tiles<<<B_ * (N_ / 256), 256, 0, stream>>>(xp, gp, rowmin, colmin);
    cd_reduce<<<1, 256, 0, stream>>>(rowmin, colmin, out);
}